// GCN_8151847928161
// MI455X (gfx1250) — compile-verified
//
#include <hip/hip_runtime.h>
#include <math.h>

#define NN 50000
#define EE 600000
#define GG 1000
#define HC 32
#define EPSV 1e-5f

typedef __attribute__((ext_vector_type(2))) float v2f;
typedef __attribute__((ext_vector_type(8))) float v8f;

static inline int cdiv(long long a, int b) { return (int)((a + b - 1) / b); }

// ---------- atomic float min/max via sign-magnitude int atomics ----------
__device__ __forceinline__ void atomicMinF(float* a, float v) {
  if (v >= 0.0f) atomicMin((int*)a, __float_as_int(v));
  else           atomicMax((unsigned int*)a, __float_as_uint(v));
}
__device__ __forceinline__ void atomicMaxF(float* a, float v) {
  if (v >= 0.0f) atomicMax((int*)a, __float_as_int(v));
  else           atomicMin((unsigned int*)a, __float_as_uint(v));
}

// ---------- WMMA fp32 GEMM: Y[N,ncol] = X[N,32] @ Weff[32,ncol] (+bias) (+=Y) ----------
// Weff[k][c] = trans ? W[c*32+k] : W[k*ncol+c].  One wave per 16-row block.
// A layout (16x4 f32): lanes 0-15 rows M=0..15 w/ K={k0,k0+1}; lanes 16-31 K={k0+2,k0+3}.
// B layout (4x16 f32): VGPR v, lanes 0-15 -> K=v col=lane; lanes 16-31 -> K=v+2.
// C/D layout (16x16 f32): VGPR v, lanes 0-15 -> M=v; lanes 16-31 -> M=v+8; N=lane&15.
__global__ void k_gemm32(const float* __restrict__ X, const float* __restrict__ W,
                         const float* __restrict__ bias, float* __restrict__ Y,
                         int ncol, int trans, int accum) {
  int rb   = blockIdx.x;
  int lane = threadIdx.x;      // 0..31, wave32
  int l15  = lane & 15;
  int kh   = lane >> 4;
  const float* xr = X + (size_t)(rb * 16 + l15) * HC + kh * 2;
  int rbase = rb * 16 + kh * 8;

  // Preload the 8 A-fragments (contiguous 8B pairs) once; reused across col tiles.
  v2f afrag[8];
  #pragma unroll
  for (int k0 = 0; k0 < 8; ++k0) afrag[k0] = *(const v2f*)(xr + k0 * 4);

  for (int ct = 0; ct < ncol; ct += 16) {
    int col = ct + l15;
    v8f acc = {};
    if (accum) {
      #pragma unroll
      for (int v = 0; v < 8; ++v) acc[v] = Y[(size_t)(rbase + v) * ncol + col];
    }
    #pragma unroll
    for (int k0 = 0; k0 < 8; ++k0) {
      int ka = k0 * 4 + kh * 2;
      v2f b;
      if (trans) {
        b = *(const v2f*)(W + col * HC + ka);      // contiguous pair, b64 load
      } else {
        b.x = W[ka * ncol + col];
        b.y = W[(ka + 1) * ncol + col];
      }
      acc = __builtin_amdgcn_wmma_f32_16x16x4_f32(false, afrag[k0], false, b, (short)0,
                                                  acc, false, false);
    }
    float bv = bias ? bias[col] : 0.0f;
    #pragma unroll
    for (int v = 0; v < 8; ++v) Y[(size_t)(rbase + v) * ncol + col] = acc[v] + bv;
  }
}

// ---------- misc elementwise ----------
__global__ void k_fill(float* p, float v, int n) {
  int t = blockIdx.x * blockDim.x + threadIdx.x;
  if (t < n) p[t] = v;
}
__global__ void k_relu(float* p, int n) {
  int t = blockIdx.x * blockDim.x + threadIdx.x;
  if (t < n) p[t] = fmaxf(p[t], 0.0f);
}
__global__ void k_elu(float* p, int n) {
  int t = blockIdx.x * blockDim.x + threadIdx.x;
  if (t < n) { float x = p[t]; p[t] = x > 0.0f ? x : expm1f(x); }
}
__global__ void k_softplus(float* p, int n) {
  int t = blockIdx.x * blockDim.x + threadIdx.x;
  if (t < n) { float x = p[t]; p[t] = fmaxf(x, 0.0f) + log1pf(expf(-fabsf(x))); }
}

// ---------- AtomEncoder: h[i,c] = sum_j emb[j, x[i,j], c] ----------
__global__ void k_atom(const int* __restrict__ x, const float* __restrict__ emb,
                       float* __restrict__ h) {
  int t = blockIdx.x * blockDim.x + threadIdx.x;
  if (t >= NN * HC) return;
  int i = t >> 5, c = t & 31;
  float s = 0.0f;
  #pragma unroll
  for (int j = 0; j < 9; ++j) {
    int xv = x[i * 9 + j];
    s += emb[(j * 100 + xv) * HC + c];
  }
  h[t] = s;
}

// ---------- degree (dst) ----------
__global__ void k_deg(const int* __restrict__ dst, float* __restrict__ deg) {
  int e = blockIdx.x * blockDim.x + threadIdx.x;
  if (e < EE) atomicAdd(&deg[dst[e]], 1.0f);
}

// ---------- edge scatters over E*32 lanes ----------
__global__ void k_scatter_min(const float* __restrict__ m, const int* __restrict__ src,
                              const int* __restrict__ dst, float* __restrict__ out) {
  int t = blockIdx.x * blockDim.x + threadIdx.x;
  if (t >= EE * HC) return;
  int e = t >> 5, c = t & 31;
  atomicMinF(&out[dst[e] * HC + c], m[src[e] * HC + c]);
}
__global__ void k_scatter_sum(const float* __restrict__ m, const int* __restrict__ src,
                              const int* __restrict__ dst, float* __restrict__ out) {
  int t = blockIdx.x * blockDim.x + threadIdx.x;
  if (t >= EE * HC) return;
  int e = t >> 5, c = t & 31;
  atomicAdd(&out[dst[e] * HC + c], m[src[e] * HC + c]);
}
__global__ void k_scatter_mul(const float* __restrict__ m, const int* __restrict__ src,
                              const int* __restrict__ dst, float* __restrict__ logs,
                              float* __restrict__ neg) {
  int t = blockIdx.x * blockDim.x + threadIdx.x;
  if (t >= EE * HC) return;
  int e = t >> 5, c = t & 31;
  float v = m[src[e] * HC + c];
  int o = dst[e] * HC + c;
  atomicAdd(&logs[o], logf(fabsf(v)));
  atomicAdd(&neg[o], v < 0.0f ? 1.0f : 0.0f);
}
__global__ void k_tag_edge(const float* __restrict__ cur, const int* __restrict__ src,
                           const int* __restrict__ dst, const float* __restrict__ deg,
                           float* __restrict__ out) {
  int t = blockIdx.x * blockDim.x + threadIdx.x;
  if (t >= EE * HC) return;
  int e = t >> 5, c = t & 31;
  int s = src[e], d = dst[e];
  float ds_ = deg[s], dd = deg[d];
  float nis = ds_ > 0.0f ? rsqrtf(fmaxf(ds_, 1.0f)) : 0.0f;
  float nid = dd > 0.0f ? rsqrtf(fmaxf(dd, 1.0f)) : 0.0f;
  atomicAdd(&out[d * HC + c], nis * nid * cur[s * HC + c]);
}
__global__ void k_sage_edge(const float* __restrict__ h, const int* __restrict__ src,
                            const int* __restrict__ dst, float* __restrict__ s1,
                            float* __restrict__ s2) {
  int t = blockIdx.x * blockDim.x + threadIdx.x;
  if (t >= EE * HC) return;
  int e = t >> 5, c = t & 31;
  float v = h[src[e] * HC + c];
  int o = dst[e] * HC + c;
  atomicAdd(&s1[o], v);
  atomicAdd(&s2[o], v * v);
}

// ---------- per-node combines ----------
__global__ void k_mask_empty(float* __restrict__ a, const float* __restrict__ deg) {
  int t = blockIdx.x * blockDim.x + threadIdx.x;
  if (t >= NN * HC) return;
  if (!(deg[t >> 5] > 0.0f)) a[t] = 0.0f;
}
__global__ void k_mul_combine(float* __restrict__ logs, const float* __restrict__ neg,
                              const float* __restrict__ deg) {
  int t = blockIdx.x * blockDim.x + threadIdx.x;
  if (t >= NN * HC) return;
  if (deg[t >> 5] > 0.0f) {
    float sg = 1.0f - 2.0f * fmodf(neg[t], 2.0f);
    logs[t] = sg * expf(logs[t]);
  } else {
    logs[t] = 0.0f;
  }
}
__global__ void k_gru(const float* __restrict__ gi, const float* __restrict__ gh,
                      float* __restrict__ h) {
  int t = blockIdx.x * blockDim.x + threadIdx.x;
  if (t >= NN * HC) return;
  int i = t >> 5, c = t & 31;
  const float* gir = gi + i * 96;
  const float* ghr = gh + i * 96;
  float r  = 1.0f / (1.0f + expf(-(gir[c]      + ghr[c])));
  float z  = 1.0f / (1.0f + expf(-(gir[32 + c] + ghr[32 + c])));
  float nn = tanhf(gir[64 + c] + r * ghr[64 + c]);
  h[t] = (1.0f - z) * nn + z * h[t];
}
__global__ void k_sage_var(float* __restrict__ s1, const float* __restrict__ s2,
                           const float* __restrict__ deg) {
  int t = blockIdx.x * blockDim.x + threadIdx.x;
  if (t >= NN * HC) return;
  float dc = fmaxf(deg[t >> 5], 1.0f);
  float mu = s1[t] / dc;
  s1[t] = s2[t] / dc - mu * mu;
}

// ---------- BatchNorm over all nodes (one block per channel) ----------
__global__ void k_bn_stats(const float* __restrict__ h, float* __restrict__ stat) {
  __shared__ float ss[256], sq[256];
  int c = blockIdx.x, tid = threadIdx.x;
  float s = 0.0f, q = 0.0f;
  for (int i = tid; i < NN; i += 256) { float v = h[i * HC + c]; s += v; q += v * v; }
  ss[tid] = s; sq[tid] = q; __syncthreads();
  for (int w = 128; w > 0; w >>= 1) {
    if (tid < w) { ss[tid] += ss[tid + w]; sq[tid] += sq[tid + w]; }
    __syncthreads();
  }
  if (tid == 0) {
    float mu = ss[0] / (float)NN;
    stat[c] = mu;
    stat[HC + c] = sq[0] / (float)NN - mu * mu;
  }
}
__global__ void k_bn_apply(float* __restrict__ h, const float* __restrict__ stat,
                           const float* __restrict__ g, const float* __restrict__ b) {
  int t = blockIdx.x * blockDim.x + threadIdx.x;
  if (t >= NN * HC) return;
  int c = t & 31;
  h[t] = (h[t] - stat[c]) * rsqrtf(stat[HC + c] + EPSV) * g[c] + b[c];
}

// ---------- per-graph instance norm + pooling ----------
__global__ void k_graph_stats(const float* __restrict__ h, const int* __restrict__ batch,
                              float* __restrict__ gsum, float* __restrict__ gsq,
                              float* __restrict__ gcnt) {
  int t = blockIdx.x * blockDim.x + threadIdx.x;
  if (t >= NN * HC) return;
  int i = t >> 5, c = t & 31;
  int b = batch[i];
  float v = h[t];
  atomicAdd(&gsum[b * HC + c], v);
  atomicAdd(&gsq[b * HC + c], v * v);
  if (c == 0) atomicAdd(&gcnt[b], 1.0f);
}
__global__ void k_instnorm(float* __restrict__ h, const int* __restrict__ batch,
                           const float* __restrict__ gsum, const float* __restrict__ gsq,
                           const float* __restrict__ gcnt) {
  int t = blockIdx.x * blockDim.x + threadIdx.x;
  if (t >= NN * HC) return;
  int i = t >> 5, c = t & 31;
  int b = batch[i];
  float cnt = fmaxf(gcnt[b], 1.0f);
  float gm = gsum[b * HC + c] / cnt;
  float gv = fmaxf(gsq[b * HC + c] / cnt - gm * gm, 0.0f);
  h[t] = (h[t] - gm) * rsqrtf(gv + EPSV);
}
__global__ void k_pool_max(const float* __restrict__ h, const int* __restrict__ batch,
                           float* __restrict__ pool) {
  int t = blockIdx.x * blockDim.x + threadIdx.x;
  if (t >= NN * HC) return;
  int i = t >> 5, c = t & 31;
  atomicMaxF(&pool[batch[i] * HC + c], h[t]);
}
__global__ void k_final(const float* __restrict__ pool, const float* __restrict__ lw,
                        const float* __restrict__ lb, float* __restrict__ out) {
  int t = blockIdx.x * blockDim.x + threadIdx.x;
  if (t >= GG * 2) return;
  int g = t >> 1, j = t & 1;
  float acc = lb[j];
  #pragma unroll
  for (int c = 0; c < HC; ++c) {
    float p = pool[g * HC + c];
    unsigned u = __float_as_uint(p);
    if ((u & 0x7F800000u) == 0x7F800000u) p = 0.0f;  // !isfinite -> 0
    acc += p * lw[c * 2 + j];
  }
  out[t] = acc;
}

extern "C" void kernel_launch(void* const* d_in, const int* in_sizes, int n_in,
                              void* d_out, int out_size, void* d_ws, size_t ws_size,
                              hipStream_t stream) {
  const int*   x        = (const int*)  d_in[0];
  const int*   ei       = (const int*)  d_in[1];
  const int*   batch    = (const int*)  d_in[2];
  const float* emb      = (const float*)d_in[3];
  const float* w1       = (const float*)d_in[4];
  const float* g1_wih   = (const float*)d_in[5];
  const float* g1_whh   = (const float*)d_in[6];
  const float* g1_bih   = (const float*)d_in[7];
  const float* g1_bhh   = (const float*)d_in[8];
  const float* rel_w    = (const float*)d_in[9];
  const float* rel_b    = (const float*)d_in[10];
  const float* root_w   = (const float*)d_in[11];
  const float* bn2_g    = (const float*)d_in[12];
  const float* bn2_b    = (const float*)d_in[13];
  const float* w3       = (const float*)d_in[14];
  const float* g3_wih   = (const float*)d_in[15];
  const float* g3_whh   = (const float*)d_in[16];
  const float* g3_bih   = (const float*)d_in[17];
  const float* g3_bhh   = (const float*)d_in[18];
  const float* lin3_w   = (const float*)d_in[19];
  const float* lin3_b   = (const float*)d_in[20];
  const float* w4       = (const float*)d_in[21];
  const float* b4       = (const float*)d_in[22];
  const float* lin4_w   = (const float*)d_in[23];
  const float* lin4_b   = (const float*)d_in[24];
  const float* sage_lw  = (const float*)d_in[25];
  const float* sage_lb  = (const float*)d_in[26];
  const float* sage_rw  = (const float*)d_in[27];
  const float* lin5_w   = (const float*)d_in[28];
  const float* lin5_b   = (const float*)d_in[29];
  const float* lin_w    = (const float*)d_in[30];
  const float* lin_b    = (const float*)d_in[31];

  const int* src = ei;
  const int* dst = ei + EE;

  float* ws = (float*)d_ws;
  size_t o = 0;
  float* bufH = ws + o; o += (size_t)NN * HC;
  float* bufM = ws + o; o += (size_t)NN * HC;
  float* aggA = ws + o; o += (size_t)NN * HC;
  float* aggB = ws + o; o += (size_t)NN * HC;
  float* cur  = ws + o; o += (size_t)NN * HC;
  float* nxt  = ws + o; o += (size_t)NN * HC;
  float* acc4 = ws + o; o += (size_t)NN * HC;
  float* tmp  = ws + o; o += (size_t)NN * HC;
  float* gi   = ws + o; o += (size_t)NN * 96;
  float* gh   = ws + o; o += (size_t)NN * 96;
  float* deg  = ws + o; o += (size_t)NN;
  float* gsum = ws + o; o += (size_t)GG * HC;
  float* gsq  = ws + o; o += (size_t)GG * HC;
  float* pool = ws + o; o += (size_t)GG * HC;
  float* gcnt = ws + o; o += (size_t)GG;
  float* stat = ws + o; o += 64;

  const int NL = NN * HC;                  // 1.6M
  const int EL = EE * HC;                  // 19.2M
  const int bNL = cdiv(NL, 256), bEL = cdiv(EL, 256);
  const int RB = NN / 16;                  // 3125 row blocks

  // ---- AtomEncoder + degree ----
  k_atom<<<bNL, 256, 0, stream>>>(x, emb, bufH);
  k_fill<<<cdiv(NN, 256), 256, 0, stream>>>(deg, 0.0f, NN);
  k_deg<<<cdiv(EE, 256), 256, 0, stream>>>(dst, deg);

  // ---- conv1: GatedGraphConv x4, aggr=min ----
  for (int it = 0; it < 4; ++it) {
    k_gemm32<<<RB, 32, 0, stream>>>(bufH, w1 + it * HC * HC, nullptr, bufM, HC, 0, 0);
    k_fill<<<bNL, 256, 0, stream>>>(aggA, INFINITY, NL);
    k_scatter_min<<<bEL, 256, 0, stream>>>(bufM, src, dst, aggA);
    k_mask_empty<<<bNL, 256, 0, stream>>>(aggA, deg);
    k_gemm32<<<RB, 32, 0, stream>>>(aggA, g1_wih, g1_bih, gi, 96, 1, 0);
    k_gemm32<<<RB, 32, 0, stream>>>(bufH, g1_whh, g1_bhh, gh, 96, 1, 0);
    k_gru<<<bNL, 256, 0, stream>>>(gi, gh, bufH);
  }
  k_relu<<<bNL, 256, 0, stream>>>(bufH, NL);

  // ---- conv2: GraphConv (sum) + elu ----
  k_fill<<<bNL, 256, 0, stream>>>(aggA, 0.0f, NL);
  k_scatter_sum<<<bEL, 256, 0, stream>>>(bufH, src, dst, aggA);
  k_gemm32<<<RB, 32, 0, stream>>>(aggA, rel_w, rel_b, bufM, HC, 0, 0);
  k_gemm32<<<RB, 32, 0, stream>>>(bufH, root_w, nullptr, bufM, HC, 0, 1);
  k_elu<<<bNL, 256, 0, stream>>>(bufM, NL);
  float* H = bufM;  float* T = bufH;   // H = features, T = spare

  // ---- bn2 ----
  k_bn_stats<<<HC, 256, 0, stream>>>(H, stat);
  k_bn_apply<<<bNL, 256, 0, stream>>>(H, stat, bn2_g, bn2_b);

  // ---- conv3: GatedGraphConv x5, aggr=mul ----
  for (int it = 0; it < 5; ++it) {
    k_gemm32<<<RB, 32, 0, stream>>>(H, w3 + it * HC * HC, nullptr, T, HC, 0, 0);
    k_fill<<<bNL, 256, 0, stream>>>(aggA, 0.0f, NL);
    k_fill<<<bNL, 256, 0, stream>>>(aggB, 0.0f, NL);
    k_scatter_mul<<<bEL, 256, 0, stream>>>(T, src, dst, aggA, aggB);
    k_mul_combine<<<bNL, 256, 0, stream>>>(aggA, aggB, deg);
    k_gemm32<<<RB, 32, 0, stream>>>(aggA, g3_wih, g3_bih, gi, 96, 1, 0);
    k_gemm32<<<RB, 32, 0, stream>>>(H, g3_whh, g3_bhh, gh, 96, 1, 0);
    k_gru<<<bNL, 256, 0, stream>>>(gi, gh, H);
  }
  k_gemm32<<<RB, 32, 0, stream>>>(H, lin3_w, lin3_b, T, HC, 0, 0);
  k_relu<<<bNL, 256, 0, stream>>>(T, NL);
  { float* s = H; H = T; T = s; }   // H=bufH, T=bufM

  // ---- conv4: TAGConv K=7 ----
  k_gemm32<<<RB, 32, 0, stream>>>(H, w4, b4, acc4, HC, 0, 0);
  {
    const float* curp = H;
    float* A = cur; float* B = nxt;
    for (int k = 1; k <= 7; ++k) {
      k_fill<<<bNL, 256, 0, stream>>>(A, 0.0f, NL);
      k_tag_edge<<<bEL, 256, 0, stream>>>(curp, src, dst, deg, A);
      k_gemm32<<<RB, 32, 0, stream>>>(A, w4 + k * HC * HC, nullptr, acc4, HC, 0, 1);
      curp = A; float* s = A; A = B; B = s;
    }
  }
  k_gemm32<<<RB, 32, 0, stream>>>(acc4, lin4_w, lin4_b, T, HC, 0, 0);
  k_softplus<<<bNL, 256, 0, stream>>>(T, NL);
  { float* s = H; H = T; T = s; }   // H=bufM, T=bufH

  // ---- conv5: SAGE with var aggregation ----
  k_fill<<<bNL, 256, 0, stream>>>(aggA, 0.0f, NL);
  k_fill<<<bNL, 256, 0, stream>>>(aggB, 0.0f, NL);
  k_sage_edge<<<bEL, 256, 0, stream>>>(H, src, dst, aggA, aggB);
  k_sage_var<<<bNL, 256, 0, stream>>>(aggA, aggB, deg);
  k_gemm32<<<RB, 32, 0, stream>>>(aggA, sage_lw, sage_lb, tmp, HC, 0, 0);
  k_gemm32<<<RB, 32, 0, stream>>>(H, sage_rw, nullptr, tmp, HC, 0, 1);
  k_gemm32<<<RB, 32, 0, stream>>>(tmp, lin5_w, lin5_b, T, HC, 0, 0);
  k_softplus<<<bNL, 256, 0, stream>>>(T, NL);
  H = T;                             // H=bufH

  // ---- bn5: per-graph instance norm ----
  k_fill<<<cdiv(GG * HC * 2, 256), 256, 0, stream>>>(gsum, 0.0f, GG * HC * 2); // gsum+gsq
  k_fill<<<cdiv(GG, 256), 256, 0, stream>>>(gcnt, 0.0f, GG);
  k_graph_stats<<<bNL, 256, 0, stream>>>(H, batch, gsum, gsq, gcnt);
  k_instnorm<<<bNL, 256, 0, stream>>>(H, batch, gsum, gsq, gcnt);

  // ---- global max pool + final linear ----
  k_fill<<<cdiv(GG * HC, 256), 256, 0, stream>>>(pool, -INFINITY, GG * HC);
  k_pool_max<<<bNL, 256, 0, stream>>>(H, batch, pool);
  k_final<<<cdiv(GG * 2, 256), 256, 0, stream>>>(pool, lin_w, lin_b, (float*)d_out);
}